// CatNet_14723147891126
// MI455X (gfx1250) — compile-verified
//
#include <hip/hip_runtime.h>

// Moments of a tridiagonal (Jacobi-type) operator:
//   out[m] = e0^T T^m (x*e0),  m = 0..2047,  T tridiag(1, c[i], g[i]), 2048x2048
// Two-sided Krylov split: mu_{j+k} = w_j . v_k  with  v_k = T^k (x e0),
// w_j = (T^T)^j e0.  Phase 1: two depth-1024 LDS-resident chains (halved
// critical path).  Phase 2: mu_{1024+k} = w_1024 . v_k = row 15 of the
// Hankel block G = W16 x V  computed with V_WMMA_F32_16X16X4_F32.

#define H 1024

typedef __attribute__((ext_vector_type(2))) float v2f;
typedef __attribute__((ext_vector_type(8))) float v8f;

// ---------------------------------------------------------------------------
// Phase 1: block 0 runs the v-chain (1023 steps), block 1 the w-chain (1024
// steps).  One thread per state element, neighbor exchange through double-
// buffered LDS, exactly one workgroup barrier per step.
// ---------------------------------------------------------------------------
__global__ __launch_bounds__(1024)
void krylov_phase1(const float* __restrict__ x,
                   const float* __restrict__ betas,
                   const float* __restrict__ gammas,
                   float* __restrict__ out,   // 2048 f32
                   float* __restrict__ V,     // H x H, column-major (col k = v_k)
                   float* __restrict__ W16)   // 16 x H, row-major (row r = w_{1009+r})
{
    __shared__ float sbuf[2][H + 2];   // +2: zero halo at both ends
    const int t = threadIdx.x;

    // Per-thread tridiagonal coefficients (registers only; no LDS needed).
    const float c   = betas[t == 0 ? 0 : t - 1];
    const float g   = gammas[t];
    const float gm1 = (t == 0) ? 0.0f : gammas[t - 1];

    if (t == 0) {  // permanent zero halos
        sbuf[0][0] = 0.0f; sbuf[0][H + 1] = 0.0f;
        sbuf[1][0] = 0.0f; sbuf[1][H + 1] = 0.0f;
    }

    if (blockIdx.x == 0) {
        // ---- v-chain: v_s[t] = v_{s-1}[t-1] + c[t]*v_{s-1}[t] + g[t]*v_{s-1}[t+1]
        float val = (t == 0) ? x[0] : 0.0f;
        sbuf[0][t + 1] = val;
        V[t] = val;                       // column 0 = x*e0
        if (t == 0) out[0] = 1.0f;        // reference hard-codes leading 1
        __syncthreads();
        for (int s = 1; s < H; ++s) {
            const float* rb = sbuf[(s + 1) & 1];
            const float l = rb[t];        // neighbor t-1 (halo-shifted)
            const float r = rb[t + 2];    // neighbor t+1
            val = l + c * val + g * r;
            sbuf[s & 1][t + 1] = val;
            V[s * H + t] = val;           // stream column v_s (coalesced)
            if (t == 0) out[s] = val;     // mu_s for s < 1024
            __syncthreads();
        }
    } else {
        // ---- w-chain: w_s[t] = g[t-1]*w_{s-1}[t-1] + c[t]*w_{s-1}[t] + w_{s-1}[t+1]
        float val = (t == 0) ? 1.0f : 0.0f;
        sbuf[0][t + 1] = val;
        __syncthreads();
        for (int s = 1; s <= H; ++s) {
            const float* rb = sbuf[(s + 1) & 1];
            const float l = rb[t];
            const float r = rb[t + 2];
            val = gm1 * l + c * val + r;
            sbuf[s & 1][t + 1] = val;
            if (s >= H - 15)              // keep w_1009 .. w_1024
                W16[(s - (H - 15)) * H + t] = val;
            __syncthreads();
        }
    }
}

// ---------------------------------------------------------------------------
// Phase 2: G = W16 (16 x 1024) x V (1024 x 1024) via V_WMMA_F32_16X16X4_F32.
// One wave per 16x16 tile of G; 64 workgroups cover all 1024 columns.
// out[1024+k] = G[15, k] = w_1024 . v_k.
// A-fragment layout (ISA 7.12.2): lane L, m = L&15, half = L>>4:
//   {A[m][i+2*half], A[m][i+2*half+1]}  -> contiguous float2 from row-major W16.
// B-fragment assumed symmetric: {B[2*half][n], B[2*half+1][n]}
//   -> contiguous float2 from column-major V.
// ---------------------------------------------------------------------------
__global__ __launch_bounds__(32)
void hankel_wmma_phase2(const float* __restrict__ V,
                        const float* __restrict__ W16,
                        float* __restrict__ out)
{
    const int lane  = threadIdx.x;
    const int half  = lane >> 4;
    const int m     = lane & 15;
    const int nbase = blockIdx.x * 16;

    const v2f* __restrict__ Ap = (const v2f*)(W16 + m * H);            // row m
    const v2f* __restrict__ Bp = (const v2f*)(V + (nbase + m) * H);    // col nbase+m

    v8f acc = {0.f, 0.f, 0.f, 0.f, 0.f, 0.f, 0.f, 0.f};
    #pragma unroll 4
    for (int i = 0; i < H; i += 4) {
        v2f a = Ap[(i >> 1) + half];   // elements i+2*half, i+2*half+1
        v2f b = Bp[(i >> 1) + half];
        acc = __builtin_amdgcn_wmma_f32_16x16x4_f32(
            /*neg_a=*/false, a, /*neg_b=*/false, b,
            /*c_mod=*/(short)0, acc, /*reuse_a=*/false, /*reuse_b=*/false);
    }

    // D layout: M=15 -> VGPR 7, lanes 16..31, N = lane-16 = m.
    const float r15 = acc[7];
    if (half == 1)
        out[1024 + nbase + m] = r15;
}

// ---------------------------------------------------------------------------
extern "C" void kernel_launch(void* const* d_in, const int* in_sizes, int n_in,
                              void* d_out, int out_size, void* d_ws, size_t ws_size,
                              hipStream_t stream) {
    const float* x      = (const float*)d_in[0];
    const float* betas  = (const float*)d_in[1];
    const float* gammas = (const float*)d_in[2];
    float* out = (float*)d_out;

    float* V   = (float*)d_ws;                  // H*H floats (4 MB)
    float* W16 = V + (size_t)H * H;             // 16*H floats (64 KB)

    krylov_phase1<<<dim3(2), dim3(1024), 0, stream>>>(x, betas, gammas, out, V, W16);
    hankel_wmma_phase2<<<dim3(H / 16), dim3(32), 0, stream>>>(V, W16, out);
}